// TreeProjector_712964571643
// MI455X (gfx1250) — compile-verified
//
#include <hip/hip_runtime.h>
#include <hip/hip_bf16.h>
#include <math.h>

#define G 320
#define NPTS 2000000
#define RADIUS 4
#define KLEN 9
#define THR_FRAC 0.1f
#define TSEG 5   // n-tiles per wave segment; 20 tiles/line = 4 segments

typedef float v2f __attribute__((ext_vector_type(2)));
typedef float v8f __attribute__((ext_vector_type(8)));

// Zero-padded normalized Gaussian taps (sigma=1, radius=4), exact f32 constants.
// PTAB[d + 15] == g[d] for d in [0,8], else 0.  d = k - n ranges over [-15, 23].
__device__ const float PTAB[40] = {
    0.f, 0.f, 0.f, 0.f, 0.f, 0.f, 0.f, 0.f, 0.f, 0.f, 0.f, 0.f, 0.f, 0.f, 0.f,
    1.3383062e-4f, 4.4318616e-3f, 5.3991128e-2f, 2.4197145e-1f, 3.9894347e-1f,
    2.4197145e-1f, 5.3991128e-2f, 4.4318616e-3f, 1.3383062e-4f,
    0.f, 0.f, 0.f, 0.f, 0.f, 0.f, 0.f, 0.f, 0.f, 0.f, 0.f, 0.f, 0.f, 0.f, 0.f, 0.f
};

// ---- order-preserving float<->uint keys for atomic min over signed floats ----
__device__ __forceinline__ unsigned fkey(float f) {
    unsigned u = __float_as_uint(f);
    return (u & 0x80000000u) ? ~u : (u | 0x80000000u);
}
__device__ __forceinline__ float fdec(unsigned k) {
    unsigned u = (k & 0x80000000u) ? (k ^ 0x80000000u) : ~k;
    return __uint_as_float(u);
}

// ---------------- K0: zero histogram grid + init atomic scalars ----------------
__global__ void k_init(float* __restrict__ gridA, unsigned* __restrict__ scal, size_t n) {
    size_t i = (size_t)blockIdx.x * blockDim.x + threadIdx.x;
    if (i == 0) { scal[0] = 0xFFFFFFFFu; scal[1] = 0xFFFFFFFFu; scal[2] = 0xFFFFFFFFu; scal[3] = 0u; }
    size_t stride = (size_t)gridDim.x * blockDim.x;
    for (size_t j = i; j < n; j += stride) gridA[j] = 0.0f;
}

// ---------------- K1: per-axis global min of votes ----------------
__global__ void k_minred(const float* __restrict__ coords, const float* __restrict__ dir,
                         const float* __restrict__ mag, unsigned* __restrict__ scal) {
    int i = blockIdx.x * blockDim.x + threadIdx.x;
    int stride = gridDim.x * blockDim.x;
    float mn0 = INFINITY, mn1 = INFINITY, mn2 = INFINITY;
    for (int p = i; p < NPTS; p += stride) {
        float e = expm1f(mag[p]);
        mn0 = fminf(mn0, coords[3 * p + 0] + dir[3 * p + 0] * e);
        mn1 = fminf(mn1, coords[3 * p + 1] + dir[3 * p + 1] * e);
        mn2 = fminf(mn2, coords[3 * p + 2] + dir[3 * p + 2] * e);
    }
    float m[3] = {mn0, mn1, mn2};
#pragma unroll
    for (int a = 0; a < 3; a++) {
        float v = m[a];
#pragma unroll
        for (int off = 16; off > 0; off >>= 1) v = fminf(v, __shfl_xor(v, off, 32));
        if ((threadIdx.x & 31) == 0) atomicMin(&scal[a], fkey(v));
    }
}

// ---------------- K2: scatter-add histogram ----------------
__global__ void k_scatter(const float* __restrict__ coords, const float* __restrict__ dir,
                          const float* __restrict__ mag, const float* __restrict__ w,
                          const unsigned* __restrict__ scal, float* __restrict__ gridA) {
    int i = blockIdx.x * blockDim.x + threadIdx.x;
    int stride = gridDim.x * blockDim.x;
    float mnx = floorf(fdec(scal[0]));
    float mny = floorf(fdec(scal[1]));
    float mnz = floorf(fdec(scal[2]));
    for (int p = i; p < NPTS; p += stride) {
        float e = expm1f(mag[p]);
        // rintf = round-half-even, matches jnp.round
        int ix = (int)rintf(coords[3 * p + 0] + dir[3 * p + 0] * e - mnx);
        int iy = (int)rintf(coords[3 * p + 1] + dir[3 * p + 1] * e - mny);
        int iz = (int)rintf(coords[3 * p + 2] + dir[3 * p + 2] * e - mnz);
        ix = min(max(ix, 0), G - 1);
        iy = min(max(iy, 0), G - 1);
        iz = min(max(iz, 0), G - 1);
        atomicAdd(&gridA[(ix * G + iy) * G + iz], w[p]);
    }
}

// ---------------- K3: one separable Gaussian pass as banded WMMA GEMM ----------------
// out[m][n] = sum_{k=0..23} In_ext[m][k] * Kband[k][n],  Kband[k][n] = g[k-n] for 0<=k-n<=8.
// Compile-time strides (immediate-offset addressing); each wave walks a TSEG-tile
// segment along the conv axis, rolling the 2 overlapping A-chunks between tiles.
// Wave-uniform control values are forced into SGPRs via readfirstlane so the
// interior/edge tile selection is a scalar s_cbranch, not an EXEC-masked region.
template <int SM, int SN, int SO>
__global__ void k_conv(const float* __restrict__ in, float* __restrict__ out,
                       int nM, int nSegs, int nO) {
    int lane = threadIdx.x & 31;
    int wib  = threadIdx.x >> 5;
    int wid  = blockIdx.x * (blockDim.x >> 5) + wib;
    int total = nM * nSegs * nO;
    if (wid >= total) return;  // whole wave exits together: EXEC all-ones for WMMA

    wid = __builtin_amdgcn_readfirstlane(wid);  // wave-uniform -> SGPR math below
    int io = wid % nO;
    int r  = wid / nO;
    int is = r % nSegs;
    int im = r / nSegs;

    int h  = lane >> 4;   // lane half: K split 2/2 per ISA A/B f32 layouts
    int lm = lane & 15;   // A row M / B,D column N

    // B chunks: local k = 4c + 2h + v, column n = lm; one padded-table load each,
    // loop-invariant -> stays hoisted in 12 VGPRs (loads are never rematerialized).
    v2f B[6];
#pragma unroll
    for (int c = 0; c < 6; c++) {
#pragma unroll
        for (int v = 0; v < 2; v++) {
            int d = (4 * c + 2 * h + v) - lm;
            B[c][v] = PTAB[d + 15];
        }
    }

    const float* rowp = in  + (long)io * SO + (long)(im * 16 + lm) * SM;  // this lane's A row
    float*       orow = out + (long)io * SO + (long)(im * 16) * SM;

    // clamped chunk load (edge tiles): always-legal address + value mask, no EXEC branch
    auto loadc = [&](int qb) -> v2f {
        v2f a;
#pragma unroll
        for (int v = 0; v < 2; v++) {
            int q  = qb + v;
            int qc = min(max(q, 0), G - 1);
            float val = rowp[(long)qc * SN];
            a[v] = (q == qc) ? val : 0.0f;
        }
        return a;
    };

    v2f a0, a1;  // rolling chunks (tile t chunks 4,5 == tile t+1 chunks 0,1)
    for (int tt = 0; tt < TSEG; tt++) {
        int t  = is * TSEG + tt;           // SGPR (is uniform, tt literal)
        int qb = 16 * t - RADIUS + 2 * h;  // q of chunk 0, this lane half
        v2f a2, a3, a4, a5;
        // uniform scalar branch: condition computed purely from SGPR t
        if (__builtin_amdgcn_readfirstlane(
                (int)((16 * t - RADIUS >= 0) && (16 * t + 19 < G)))) {
            const float* p = rowp + (long)qb * SN;   // constant immediate offsets below
            if (tt == 0) {
                a0[0] = p[0 * SN];  a0[1] = p[1 * SN];
                a1[0] = p[4 * SN];  a1[1] = p[5 * SN];
            }
            a2[0] = p[ 8 * SN]; a2[1] = p[ 9 * SN];
            a3[0] = p[12 * SN]; a3[1] = p[13 * SN];
            a4[0] = p[16 * SN]; a4[1] = p[17 * SN];
            a5[0] = p[20 * SN]; a5[1] = p[21 * SN];
        } else {
            if (tt == 0) { a0 = loadc(qb); a1 = loadc(qb + 4); }
            a2 = loadc(qb + 8);
            a3 = loadc(qb + 12);
            a4 = loadc(qb + 16);
            a5 = loadc(qb + 20);
        }

        v8f acc = {0.f, 0.f, 0.f, 0.f, 0.f, 0.f, 0.f, 0.f};
        acc = __builtin_amdgcn_wmma_f32_16x16x4_f32(false, a0, false, B[0], (short)0, acc, false, false);
        acc = __builtin_amdgcn_wmma_f32_16x16x4_f32(false, a1, false, B[1], (short)0, acc, false, false);
        acc = __builtin_amdgcn_wmma_f32_16x16x4_f32(false, a2, false, B[2], (short)0, acc, false, false);
        acc = __builtin_amdgcn_wmma_f32_16x16x4_f32(false, a3, false, B[3], (short)0, acc, false, false);
        acc = __builtin_amdgcn_wmma_f32_16x16x4_f32(false, a4, false, B[4], (short)0, acc, false, false);
        acc = __builtin_amdgcn_wmma_f32_16x16x4_f32(false, a5, false, B[5], (short)0, acc, false, false);

        // D layout: VGPR v -> M = v + 8*h, N = lm  (tile column n = 16t + lm)
        if constexpr (SM == 1) {
            *(v8f*)(orow + 8 * h + (long)(16 * t + lm) * SN) = acc;  // 32B-aligned
        } else {
#pragma unroll
            for (int v = 0; v < 8; v++)
                orow[(long)(v + 8 * h) * SM + (long)(16 * t + lm) * SN] = acc[v];
        }

        a0 = a4;  // roll overlap into next tile
        a1 = a5;
    }
}

// ---------------- K4: global max of smoothed grid (values >= 0) ----------------
__global__ void k_maxred(const float* __restrict__ a, unsigned* __restrict__ scal, size_t n) {
    size_t i = (size_t)blockIdx.x * blockDim.x + threadIdx.x;
    size_t stride = (size_t)gridDim.x * blockDim.x;
    float m = 0.0f;
    for (size_t j = i; j < n; j += stride) m = fmaxf(m, a[j]);
#pragma unroll
    for (int off = 16; off > 0; off >>= 1) m = fmaxf(m, __shfl_xor(m, off, 32));
    if ((threadIdx.x & 31) == 0) atomicMax(&scal[3], __float_as_uint(m));  // monotonic for f >= 0
}

// ---------------- K5: 3x3x3 NMS + threshold ----------------
// clamp-to-edge max == -inf-padded max (clamped duplicates are already in the
// valid neighborhood and duplicates never change a max) -> 27 unconditional loads.
__global__ void k_peaks(const float* __restrict__ acc, const unsigned* __restrict__ scal,
                        float* __restrict__ peaks) {
    size_t i = (size_t)blockIdx.x * blockDim.x + threadIdx.x;
    size_t n = (size_t)G * G * G;
    if (i >= n) return;
    int z = (int)(i % G);
    int y = (int)((i / G) % G);
    int x = (int)(i / ((size_t)G * G));
    float a = acc[i];
    float mx = -INFINITY;
#pragma unroll
    for (int dx = -1; dx <= 1; dx++) {
        int xx = min(max(x + dx, 0), G - 1);
#pragma unroll
        for (int dy = -1; dy <= 1; dy++) {
            int yy = min(max(y + dy, 0), G - 1);
#pragma unroll
            for (int dz = -1; dz <= 1; dz++) {
                int zz = min(max(z + dz, 0), G - 1);
                mx = fmaxf(mx, acc[((size_t)xx * G + yy) * G + zz]);
            }
        }
    }
    float thr = THR_FRAC * __uint_as_float(scal[3]);
    peaks[i] = (a == mx && a > thr) ? 1.0f : 0.0f;
}

extern "C" void kernel_launch(void* const* d_in, const int* in_sizes, int n_in,
                              void* d_out, int out_size, void* d_ws, size_t ws_size,
                              hipStream_t stream) {
    const float* coords = (const float*)d_in[0];  // [N,3]
    const float* dir    = (const float*)d_in[1];  // [N,3]
    const float* mag    = (const float*)d_in[2];  // [N,1]
    const float* w      = (const float*)d_in[3];  // [N]
    (void)in_sizes; (void)n_in; (void)out_size; (void)ws_size;

    const size_t Ncells = (size_t)G * G * G;
    float* GB = (float*)d_out;          // final accum lives here
    float* GA = GB + Ncells;            // scratch grid; finally overwritten with peaks
    unsigned* scal = (unsigned*)d_ws;   // [min_x, min_y, min_z, max] keys (16 bytes)

    k_init   <<<4096, 256, 0, stream>>>(GA, scal, Ncells);
    k_minred <<<2048, 256, 0, stream>>>(coords, dir, mag, scal);
    k_scatter<<<4096, 256, 0, stream>>>(coords, dir, mag, w, scal, GA);

    // 20 M-tiles * 4 segments * 320 planes = 25600 waves -> 3200 blocks of 8 waves
    const int TPB = 256, NB = 3200;
    const int nM = G / 16, nSegs = (G / 16) / TSEG, nO = G;
    // pass 1: conv along z (SN=1); M along y (SM=G); planes over x (SO=G*G)
    k_conv<G, 1, G * G><<<NB, TPB, 0, stream>>>(GA, GB, nM, nSegs, nO);
    // pass 2: conv along y (SN=G); M along z (SM=1, v8f stores); planes over x
    k_conv<1, G, G * G><<<NB, TPB, 0, stream>>>(GB, GA, nM, nSegs, nO);
    // pass 3: conv along x (SN=G*G); M along z (SM=1); planes over y (SO=G)
    k_conv<1, G * G, G><<<NB, TPB, 0, stream>>>(GA, GB, nM, nSegs, nO);

    k_maxred<<<4096, 256, 0, stream>>>(GB, scal, Ncells);
    k_peaks <<<(int)((Ncells + 255) / 256), 256, 0, stream>>>(GB, scal, GA);
}